// RGCN_28346784153940
// MI455X (gfx1250) — compile-verified
//
#include <hip/hip_runtime.h>

#define DIM 128
#define NREL 4
#define NLAYERS 3

typedef float v2f __attribute__((ext_vector_type(2)));
typedef float v8f __attribute__((ext_vector_type(8)));

// ---------------- utility: zero a buffer (float4 granularity) ----------------
__global__ __launch_bounds__(256) void zero_f4(float* __restrict__ p, long n4) {
    long i = (long)blockIdx.x * 256 + threadIdx.x;
    if (i < n4) ((float4*)p)[i] = make_float4(0.f, 0.f, 0.f, 0.f);
}

// ---------------- per-(relation,dst) edge counts -----------------------------
__global__ __launch_bounds__(256)
void rgcn_count(const int* __restrict__ dst, const int* __restrict__ etype,
                float* __restrict__ cnt, int nEdges, int nNodes) {
    int e = blockIdx.x * 256 + threadIdx.x;
    if (e < nEdges) {
        int r = etype[e];
        int d = dst[e];
        atomicAdd(&cnt[(size_t)r * nNodes + d], 1.0f);
    }
}

__global__ __launch_bounds__(256)
void rgcn_inv(const float* __restrict__ cnt, float* __restrict__ inv, int n) {
    int i = blockIdx.x * 256 + threadIdx.x;
    if (i < n) inv[i] = 1.0f / fmaxf(cnt[i], 1.0f);
}

// ---------------- transpose a batch of 128x128 matrices ----------------------
// out[m][n][k] = in[m][k][n]; one matrix per block, 32x32 LDS tiles.
__global__ __launch_bounds__(256)
void transpose128(const float* __restrict__ in, float* __restrict__ out, int nMat) {
    int m = blockIdx.x;
    if (m >= nMat) return;
    const float* A = in + (size_t)m * DIM * DIM;
    float* B = out + (size_t)m * DIM * DIM;
    __shared__ float tile[32][33];
    for (int t = 0; t < 16; ++t) {
        int tr = (t >> 2) * 32;
        int tc = (t & 3) * 32;
        for (int i = threadIdx.x; i < 1024; i += 256) {
            int r = i >> 5, c = i & 31;
            tile[r][c] = A[(size_t)(tr + r) * DIM + tc + c];
        }
        __syncthreads();
        for (int i = threadIdx.x; i < 1024; i += 256) {
            int r = i >> 5, c = i & 31;
            B[(size_t)(tc + r) * DIM + tr + c] = tile[c][r];
        }
        __syncthreads();
    }
}

// ---------------- scatter: agg[r][dst] += x[src]  (1 wave per edge) ----------
__global__ __launch_bounds__(256)
void rgcn_scatter(const int* __restrict__ src, const int* __restrict__ dst,
                  const int* __restrict__ etype, const float* __restrict__ x,
                  float* __restrict__ agg, int nEdges, int nNodes) {
    int e = blockIdx.x * 8 + (threadIdx.x >> 5);
    if (e >= nEdges) return;
    int lane = threadIdx.x & 31;
    int s = src[e];
    int d = dst[e];
    int r = etype[e];
    const float4 v = *(const float4*)(x + (size_t)s * DIM + lane * 4);
    float* out = agg + ((size_t)r * nNodes + d) * DIM + lane * 4;
    atomicAdd(out + 0, v.x);
    atomicAdd(out + 1, v.y);
    atomicAdd(out + 2, v.z);
    atomicAdd(out + 3, v.w);
}

// ---------------- fused RGCN layer GEMM via V_WMMA_F32_16X16X4_F32 -----------
// Per workgroup: 32 nodes (two 16-row tiles). A staged per source in LDS
// (17 KB, padded stride 132 -> conflict-free column reads). 8 waves each own
// one 16-col output slice; each B fragment (b64 from transposed W) feeds two
// WMMAs (row tiles 0-15 / 16-31). Effective K = 5*128 = 640 per tile.
__global__ __launch_bounds__(256)
void rgcn_gemm(const float* __restrict__ xin,    // [N,128]
               const float* __restrict__ agg,    // [4,N,128]
               const float* __restrict__ inv,    // [4,N]
               const float* __restrict__ rootT,  // [128,128] transposed (n,k)
               const float* __restrict__ WrelT,  // [4,128,128] transposed (n,k)
               const float* __restrict__ bias,   // [128]
               float* __restrict__ xout,         // [N,128]
               int nNodes, int doRelu) {
    __shared__ float As[32][132];

    const int row0 = blockIdx.x * 32;
    const int tid = threadIdx.x;
    const int wave = tid >> 5;
    const int lane = tid & 31;
    const int jbase = wave * 16;
    const int half = lane >> 4;   // 0: K pair {k,k+1}; 1: {k+2,k+3}
    const int l16 = lane & 15;
    const int col = jbase + l16;

    v8f c0 = {};
    v8f c1 = {};

    for (int s = 0; s < 5; ++s) {
        // ---- stage A for source s: 32 rows x 128 cols, 4 float4 per thread
        for (int i = tid * 4; i < 32 * DIM; i += 256 * 4) {
            int m = i >> 7;
            int d = i & 127;
            int node = row0 + m;
            float4 v = make_float4(0.f, 0.f, 0.f, 0.f);
            if (node < nNodes) {
                if (s == 0) {
                    v = *(const float4*)(xin + (size_t)node * DIM + d);
                } else {
                    int r = s - 1;
                    float sc = inv[(size_t)r * nNodes + node];
                    float4 t = *(const float4*)(agg + ((size_t)r * nNodes + node) * DIM + d);
                    v.x = t.x * sc; v.y = t.y * sc; v.z = t.z * sc; v.w = t.w * sc;
                }
            }
            *(float4*)&As[m][d] = v;
        }
        __syncthreads();

        const float* __restrict__ WT =
            (s == 0) ? rootT : (WrelT + (size_t)(s - 1) * DIM * DIM);
        const float* __restrict__ wp = WT + (size_t)col * DIM;  // column, k-major

        for (int k = 0; k < DIM; k += 4) {
            const int ka = k + half * 2;
            // B fragment: contiguous pair from transposed weights -> b64 load
            v2f b = *(const v2f*)(wp + ka);
            // A fragments (ISA 7.12.2): lane holds (A[m][ka], A[m][ka+1])
            v2f a0, a1;
            a0.x = As[l16][ka];
            a0.y = As[l16][ka + 1];
            a1.x = As[16 + l16][ka];
            a1.y = As[16 + l16][ka + 1];
            c0 = __builtin_amdgcn_wmma_f32_16x16x4_f32(
                false, a0, false, b, (short)0, c0, false, false);
            c1 = __builtin_amdgcn_wmma_f32_16x16x4_f32(
                false, a1, false, b, (short)0, c1, false, false);
        }
        __syncthreads();
    }

    // C layout: VGPR v, lanes 0-15 -> M=v, lanes 16-31 -> M=v+8; N=l16.
    const float bj = bias[col];
#pragma unroll
    for (int v = 0; v < 8; ++v) {
        int m = v + half * 8;
        int node0 = row0 + m;
        if (node0 < nNodes) {
            float val = c0[v] + bj;
            if (doRelu) val = fmaxf(val, 0.f);
            xout[(size_t)node0 * DIM + col] = val;
        }
        int node1 = row0 + 16 + m;
        if (node1 < nNodes) {
            float val = c1[v] + bj;
            if (doRelu) val = fmaxf(val, 0.f);
            xout[(size_t)node1 * DIM + col] = val;
        }
    }
}

// ---------------- host-side orchestration ------------------------------------
extern "C" void kernel_launch(void* const* d_in, const int* in_sizes, int n_in,
                              void* d_out, int out_size, void* d_ws, size_t ws_size,
                              hipStream_t stream) {
    const int* edge_index = (const int*)d_in[0];   // [2, E]
    const int* edge_type  = (const int*)d_in[1];   // [E]
    const float* x_init   = (const float*)d_in[2]; // [N, 128]
    const float* weights  = (const float*)d_in[3]; // [3, 4, 128, 128]
    const float* roots    = (const float*)d_in[4]; // [3, 128, 128]
    const float* biases   = (const float*)d_in[5]; // [3, 128]

    const int nEdges = in_sizes[1];
    const int nNodes = in_sizes[2] / DIM;
    const int* src = edge_index;
    const int* dst = edge_index + nEdges;

    // Workspace layout (floats):
    float* ws = (float*)d_ws;
    const size_t aggN = (size_t)NREL * nNodes * DIM;
    float* agg  = ws;  ws += aggN;
    float* cnt  = ws;  ws += (size_t)NREL * nNodes;
    float* inv  = ws;  ws += (size_t)NREL * nNodes;
    float* buf0 = ws;  ws += (size_t)nNodes * DIM;
    float* buf1 = ws;  ws += (size_t)nNodes * DIM;
    float* wT   = ws;  ws += (size_t)NLAYERS * NREL * DIM * DIM;
    float* rT   = ws;

    // One-time: per-(rel,dst) counts -> reciprocals (layer-invariant).
    {
        long n4 = ((long)NREL * nNodes + 3) / 4;
        zero_f4<<<(int)((n4 + 255) / 256), 256, 0, stream>>>(cnt, n4);
        rgcn_count<<<(nEdges + 255) / 256, 256, 0, stream>>>(dst, edge_type, cnt,
                                                             nEdges, nNodes);
        int nInv = NREL * nNodes;
        rgcn_inv<<<(nInv + 255) / 256, 256, 0, stream>>>(cnt, inv, nInv);
    }

    // One-time: transpose all weight matrices to (n,k) for b64 B-fragment loads.
    transpose128<<<NLAYERS * NREL, 256, 0, stream>>>(weights, wT, NLAYERS * NREL);
    transpose128<<<NLAYERS, 256, 0, stream>>>(roots, rT, NLAYERS);

    const float* xcur = x_init;
    float* outs[NLAYERS] = {buf0, buf1, (float*)d_out};
    for (int l = 0; l < NLAYERS; ++l) {
        long n4 = (long)(aggN / 4);
        zero_f4<<<(int)((n4 + 255) / 256), 256, 0, stream>>>(agg, n4);
        rgcn_scatter<<<(nEdges + 7) / 8, 256, 0, stream>>>(src, dst, edge_type,
                                                           xcur, agg, nEdges, nNodes);
        rgcn_gemm<<<(nNodes + 31) / 32, 256, 0, stream>>>(
            xcur, agg, inv,
            rT + (size_t)l * DIM * DIM,
            wT + (size_t)l * NREL * DIM * DIM,
            biases + (size_t)l * DIM,
            outs[l], nNodes, (l < NLAYERS - 1) ? 1 : 0);
        xcur = outs[l];
    }
}